// GATReg_80814104641846
// MI455X (gfx1250) — compile-verified
//
#include <hip/hip_runtime.h>

// ---------------- problem constants (from reference) ----------------
#define NNODES 50000
#define NEDGES 800000
#define EDTOT  (NNODES + NEDGES)   // 850000 edges incl. self loops
#define HEADS  4
#define HIDC   64
#define HCDIM  (HEADS * HIDC)      // 256
#define NEG_SLOPE 0.2f
#define BN_EPS 1e-5f

// GEMM tiling: 50000 rows = 16 * 3125 = (16*5) * 625 -> no tails anywhere
#define MT_PER_WAVE 5
static_assert(NNODES % (16 * MT_PER_WAVE) == 0, "M tile layout must be exact");

typedef float v2f __attribute__((ext_vector_type(2)));
typedef float v8f __attribute__((ext_vector_type(8)));

// edge_index is [2,E] row-major; self loops appended logically (src=dst=i)
__device__ __forceinline__ int edge_src(const int* __restrict__ ei, int e) {
  return (e < NEDGES) ? ei[e] : (e - NEDGES);
}
__device__ __forceinline__ int edge_dst(const int* __restrict__ ei, int e) {
  return (e < NEDGES) ? ei[NEDGES + e] : (e - NEDGES);
}

// float atomic max via signed/unsigned integer atomics (works with -inf init)
__device__ __forceinline__ void atomicMaxF32(float* addr, float v) {
  if (v >= 0.0f) atomicMax((int*)addr, __float_as_int(v));
  else           atomicMin((unsigned int*)addr, __float_as_uint(v));
}

__global__ void fill_f32(float* __restrict__ p, float v, long long n) {
  for (long long i = blockIdx.x * (long long)blockDim.x + threadIdx.x; i < n;
       i += (long long)gridDim.x * blockDim.x)
    p[i] = v;
}

// ---------------- fp32 WMMA GEMM: Out[M,NC] = X[M,K] @ W[K,NC] ----------------
// block = 128 threads (4 waves). Each wave computes MT_PER_WAVE 16x16 D tiles
// in the same N column (B fragment reused across 5 WMMAs). All offsets are
// compile-time immediates; A loads are 8B-vector loads; no predication.
// grid = (Mtiles/MT_PER_WAVE, NC/64).
template <int K, int NC>
__global__ void gemm_wmma_f32(const float* __restrict__ X, const float* __restrict__ W,
                              float* __restrict__ Out) {
  const int lane   = threadIdx.x & 31;
  const int wave   = threadIdx.x >> 5;
  const int tileM0 = blockIdx.x * MT_PER_WAVE;
  const int tileN  = blockIdx.y * 4 + wave;
  const int rc     = lane & 15;        // A row / B,D col within tile
  const int khalf  = (lane >> 4) * 2;  // lanes 0-15 -> k{0,1}, lanes 16-31 -> k{2,3}
  const int col    = tileN * 16 + rc;

  // per-lane base pointers; the unrolled k-loop indexes them with constants
  const float* __restrict__ bp = W + khalf * NC + col;
  const float* __restrict__ ap[MT_PER_WAVE];
#pragma unroll
  for (int t = 0; t < MT_PER_WAVE; ++t)
    ap[t] = X + (long long)((tileM0 + t) * 16 + rc) * K + khalf;

  v8f acc[MT_PER_WAVE];
#pragma unroll
  for (int t = 0; t < MT_PER_WAVE; ++t) acc[t] = (v8f){};

#pragma unroll
  for (int k = 0; k < K; k += 4) {
    v2f b;
    b.x = bp[k * NC];
    b.y = bp[(k + 1) * NC];
#pragma unroll
    for (int t = 0; t < MT_PER_WAVE; ++t) {
      const v2f a = *reinterpret_cast<const v2f*>(ap[t] + k);  // 8B aligned
      acc[t] = __builtin_amdgcn_wmma_f32_16x16x4_f32(false, a, false, b, (short)0,
                                                     acc[t], false, false);
    }
  }

  // D layout: VGPR r -> M = r (lanes 0-15) / M = 8+r (lanes 16-31), N = lane%16
  const int msub = (lane >> 4) * 8;
#pragma unroll
  for (int t = 0; t < MT_PER_WAVE; ++t) {
    const long long mbase = (long long)((tileM0 + t) * 16 + msub);
#pragma unroll
    for (int r = 0; r < 8; ++r)
      Out[(mbase + r) * NC + col] = acc[t][r];
  }
}

// alpha_src[n,h] = sum_c h[n,h,c]*aS[h,c] ; alpha_dst likewise. 1 thread per (n,h)
__global__ void node_alpha(const float* __restrict__ h, const float* __restrict__ aS,
                           const float* __restrict__ aD, float* __restrict__ asrc,
                           float* __restrict__ adst, int n, int H, int C) {
  int i = blockIdx.x * blockDim.x + threadIdx.x;
  if (i >= n * H) return;
  const int node = i / H, hh = i % H;
  const float* __restrict__ hp = h + (long long)node * H * C + hh * C;
  const float* __restrict__ sp = aS + hh * C;
  const float* __restrict__ dp = aD + hh * C;
  float s0 = 0.0f, s1 = 0.0f;
  for (int c = 0; c < C; c += 4) {
    const float4 hv = *reinterpret_cast<const float4*>(hp + c);
    const float4 sv = *reinterpret_cast<const float4*>(sp + c);
    const float4 dv = *reinterpret_cast<const float4*>(dp + c);
    s0 += hv.x * sv.x + hv.y * sv.y + hv.z * sv.z + hv.w * sv.w;
    s1 += hv.x * dv.x + hv.y * dv.y + hv.z * dv.z + hv.w * dv.w;
  }
  asrc[i] = s0;
  adst[i] = s1;
}

// e = leaky_relu(asrc[src]+adst[dst]); store raw score, atomic max into mmax[dst,h]
__global__ void edge_score_max(const int* __restrict__ ei, int H,
                               const float* __restrict__ asrc,
                               const float* __restrict__ adst,
                               float* __restrict__ evals, float* __restrict__ mmax,
                               long long total) {
  for (long long i = blockIdx.x * (long long)blockDim.x + threadIdx.x; i < total;
       i += (long long)gridDim.x * blockDim.x) {
    const int e = (int)(i / H), hh = (int)(i % H);
    const int s = edge_src(ei, e), d = edge_dst(ei, e);
    float v = asrc[s * H + hh] + adst[d * H + hh];
    v = (v > 0.0f) ? v : v * NEG_SLOPE;
    evals[i] = v;
    atomicMaxF32(&mmax[d * H + hh], v);
  }
}

// e <- exp(e - max[dst]); atomic add into denom[dst,h]
__global__ void edge_exp_sum(const int* __restrict__ ei, int H,
                             float* __restrict__ evals, const float* __restrict__ mmax,
                             float* __restrict__ dnm, long long total) {
  for (long long i = blockIdx.x * (long long)blockDim.x + threadIdx.x; i < total;
       i += (long long)gridDim.x * blockDim.x) {
    const int e = (int)(i / H), hh = (int)(i % H);
    const int d = edge_dst(ei, e);
    const float w = expf(evals[i] - mmax[d * H + hh]);
    evals[i] = w;
    atomicAdd(&dnm[d * H + hh], w);
  }
}

// out[dst, h, c] += h[src, h, c] * e[edge,h] / denom[dst,h]
// VEC channels per thread (VEC=4 for the 256-wide layers: b128 gather + 4 atomics)
template <int VEC>
__global__ void edge_aggregate(const int* __restrict__ ei, int H, int C,
                               const float* __restrict__ hbuf,
                               const float* __restrict__ evals,
                               const float* __restrict__ dnm,
                               float* __restrict__ out, long long total) {
  const int HC = H * C;
  const int HCv = HC / VEC;
  for (long long i = blockIdx.x * (long long)blockDim.x + threadIdx.x; i < total;
       i += (long long)gridDim.x * blockDim.x) {
    const long long e = i / HCv;
    const int r = (int)(i % HCv) * VEC;
    const int hh = r / C;  // VEC consecutive channels stay within one head (C%VEC==0)
    const int s = edge_src(ei, (int)e), d = edge_dst(ei, (int)e);
    const float alpha = evals[e * H + hh] / dnm[d * H + hh];
    const float* __restrict__ hp = hbuf + (long long)s * HC + r;
    float* __restrict__ op = out + (long long)d * HC + r;
    if (VEC == 4) {
      const float4 v = *reinterpret_cast<const float4*>(hp);  // 16B aligned
      atomicAdd(op + 0, v.x * alpha);
      atomicAdd(op + 1, v.y * alpha);
      atomicAdd(op + 2, v.z * alpha);
      atomicAdd(op + 3, v.w * alpha);
    } else {
      atomicAdd(op, hp[0] * alpha);
    }
  }
}

// t = elu(bn(agg + bias)) [+ res]; elementwise, optionally in place
__global__ void post_act(const float* __restrict__ agg, const float* __restrict__ b,
                         const float* __restrict__ g, const float* __restrict__ be,
                         const float* __restrict__ mu, const float* __restrict__ var,
                         const float* __restrict__ res, float* __restrict__ out,
                         long long n, int HC) {
  for (long long i = blockIdx.x * (long long)blockDim.x + threadIdx.x; i < n * HC;
       i += (long long)gridDim.x * blockDim.x) {
    const int c = (int)(i % HC);
    float t = agg[i] + b[c];
    t = (t - mu[c]) * (g[c] * rsqrtf(var[c] + BN_EPS)) + be[c];
    t = (t > 0.0f) ? t : expm1f(t);
    if (res) t += res[i];
    out[i] = t;
  }
}

// layer-2 matvec (K=256 -> 1) + scalar attention projections; one wave per node
__global__ void matvec_l2(const float* __restrict__ hin, const float* __restrict__ W2,
                          const float* __restrict__ as2p, const float* __restrict__ ad2p,
                          float* __restrict__ h2, float* __restrict__ as2v,
                          float* __restrict__ ad2v, int n, int K) {
  const int node = blockIdx.x * (blockDim.x >> 5) + (threadIdx.x >> 5);
  const int lane = threadIdx.x & 31;
  if (node >= n) return;
  const float* __restrict__ row = hin + (long long)node * K;
  float s = 0.0f;
  for (int k = lane; k < K; k += 32) s += row[k] * W2[k];
#pragma unroll
  for (int off = 16; off > 0; off >>= 1) s += __shfl_xor(s, off, 32);
  if (lane == 0) {
    h2[node]   = s;
    as2v[node] = s * as2p[0];
    ad2v[node] = s * ad2p[0];
  }
}

__global__ void final_bias(const float* __restrict__ agg, const float* __restrict__ b2,
                           float* __restrict__ out, int n) {
  const int i = blockIdx.x * blockDim.x + threadIdx.x;
  if (i < n) out[i] = agg[i] + b2[0];
}

static inline unsigned blks(long long total, int bs) {
  return (unsigned)((total + bs - 1) / bs);
}

extern "C" void kernel_launch(void* const* d_in, const int* in_sizes, int n_in,
                              void* d_out, int out_size, void* d_ws, size_t ws_size,
                              hipStream_t stream) {
  const float* x   = (const float*)d_in[0];
  const int*   ei  = (const int*)d_in[1];
  const float* W0  = (const float*)d_in[2];
  const float* as0 = (const float*)d_in[3];
  const float* ad0 = (const float*)d_in[4];
  const float* b0  = (const float*)d_in[5];
  const float* W1  = (const float*)d_in[6];
  const float* as1 = (const float*)d_in[7];
  const float* ad1 = (const float*)d_in[8];
  const float* b1  = (const float*)d_in[9];
  const float* W2  = (const float*)d_in[10];
  const float* as2 = (const float*)d_in[11];
  const float* ad2 = (const float*)d_in[12];
  const float* b2  = (const float*)d_in[13];
  const float* g0  = (const float*)d_in[14];
  const float* be0 = (const float*)d_in[15];
  const float* m0  = (const float*)d_in[16];
  const float* v0  = (const float*)d_in[17];
  const float* g1  = (const float*)d_in[18];
  const float* be1 = (const float*)d_in[19];
  const float* m1  = (const float*)d_in[20];
  const float* v1  = (const float*)d_in[21];
  float* out = (float*)d_out;

  // ---- workspace carve-up (~173 MB of fp32) ----
  float* p = (float*)d_ws;
  auto take = [&](long long nelem) { float* r = p; p += nelem; return r; };
  float* bufA  = take((long long)NNODES * HCDIM);  // GEMM output h (pre-agg)
  float* bufB  = take((long long)NNODES * HCDIM);  // edge-aggregate accumulator
  float* bufC  = take((long long)NNODES * HCDIM);  // layer-0 activation (residual)
  float* asrc  = take((long long)NNODES * HEADS);
  float* adst  = take((long long)NNODES * HEADS);
  float* mmax  = take((long long)NNODES * HEADS);
  float* dnm   = take((long long)NNODES * HEADS);
  float* evals = take((long long)EDTOT * HEADS);
  float* h2    = take(NNODES);
  float* as2v  = take(NNODES);
  float* ad2v  = take(NNODES);
  float* agg2  = take(NNODES);

  const long long nh   = (long long)NNODES * HEADS;
  const long long eh   = (long long)EDTOT * HEADS;
  const long long ehc4 = (long long)EDTOT * (HCDIM / 4);
  const long long nhc  = (long long)NNODES * HCDIM;
  const dim3 gemmGrid(NNODES / (16 * MT_PER_WAVE), HCDIM / 64);

  // =================== layer 0: 128 -> 4x64, concat ===================
  gemm_wmma_f32<128, HCDIM><<<gemmGrid, 128, 0, stream>>>(x, W0, bufA);
  node_alpha<<<blks(nh, 256), 256, 0, stream>>>(bufA, as0, ad0, asrc, adst,
                                                NNODES, HEADS, HIDC);
  fill_f32<<<blks(nh, 256), 256, 0, stream>>>(mmax, -__builtin_huge_valf(), nh);
  fill_f32<<<blks(nh, 256), 256, 0, stream>>>(dnm, 0.0f, nh);
  fill_f32<<<blks(nhc, 256), 256, 0, stream>>>(bufB, 0.0f, nhc);
  edge_score_max<<<blks(eh, 256), 256, 0, stream>>>(ei, HEADS, asrc, adst, evals,
                                                    mmax, eh);
  edge_exp_sum<<<blks(eh, 256), 256, 0, stream>>>(ei, HEADS, evals, mmax, dnm, eh);
  edge_aggregate<4><<<blks(ehc4, 256), 256, 0, stream>>>(ei, HEADS, HIDC, bufA,
                                                         evals, dnm, bufB, ehc4);
  post_act<<<blks(nhc, 256), 256, 0, stream>>>(bufB, b0, g0, be0, m0, v0,
                                               (const float*)nullptr, bufC,
                                               NNODES, HCDIM);

  // =================== layer 1: 256 -> 4x64, concat, +residual ===================
  gemm_wmma_f32<256, HCDIM><<<gemmGrid, 128, 0, stream>>>(bufC, W1, bufA);
  node_alpha<<<blks(nh, 256), 256, 0, stream>>>(bufA, as1, ad1, asrc, adst,
                                                NNODES, HEADS, HIDC);
  fill_f32<<<blks(nh, 256), 256, 0, stream>>>(mmax, -__builtin_huge_valf(), nh);
  fill_f32<<<blks(nh, 256), 256, 0, stream>>>(dnm, 0.0f, nh);
  fill_f32<<<blks(nhc, 256), 256, 0, stream>>>(bufB, 0.0f, nhc);
  edge_score_max<<<blks(eh, 256), 256, 0, stream>>>(ei, HEADS, asrc, adst, evals,
                                                    mmax, eh);
  edge_exp_sum<<<blks(eh, 256), 256, 0, stream>>>(ei, HEADS, evals, mmax, dnm, eh);
  edge_aggregate<4><<<blks(ehc4, 256), 256, 0, stream>>>(ei, HEADS, HIDC, bufA,
                                                         evals, dnm, bufB, ehc4);
  post_act<<<blks(nhc, 256), 256, 0, stream>>>(bufB, b1, g1, be1, m1, v1,
                                               /*res=*/bufC, bufB, NNODES, HCDIM);

  // =================== layer 2: 256 -> 1, heads=1 ===================
  matvec_l2<<<blks(NNODES, 8), 256, 0, stream>>>(bufB, W2, as2, ad2, h2, as2v,
                                                 ad2v, NNODES, HCDIM);
  fill_f32<<<blks(NNODES, 256), 256, 0, stream>>>(mmax, -__builtin_huge_valf(), NNODES);
  fill_f32<<<blks(NNODES, 256), 256, 0, stream>>>(dnm, 0.0f, NNODES);
  fill_f32<<<blks(NNODES, 256), 256, 0, stream>>>(agg2, 0.0f, NNODES);
  edge_score_max<<<blks(EDTOT, 256), 256, 0, stream>>>(ei, 1, as2v, ad2v, evals,
                                                       mmax, EDTOT);
  edge_exp_sum<<<blks(EDTOT, 256), 256, 0, stream>>>(ei, 1, evals, mmax, dnm, EDTOT);
  edge_aggregate<1><<<blks(EDTOT, 256), 256, 0, stream>>>(ei, 1, 1, h2, evals, dnm,
                                                          agg2, EDTOT);
  final_bias<<<blks(NNODES, 256), 256, 0, stream>>>(agg2, b2, out, NNODES);
}